// SignatureCalculator_20126216749981
// MI455X (gfx1250) — compile-verified
//
#include <hip/hip_runtime.h>
#include <stdint.h>
#include <math.h>

// Trajectory signature statistics, MI455X (gfx1250).
// Memory-bound streaming reduction: 201 MB in -> 80 KB out, ~8.6 us HBM floor.
// Data path: CDNA5 async global->LDS B128 copies (ASYNCcnt), NT cache hint
// (input > 192MB L2, single touch), compute from LDS, wave32 shuffle reduce.

#define TSTEPS 2048
#define NCH    6
#define BLOCK  256
#define FLOATS_PER_TRAJ (TSTEPS * NCH)                 // 12288 floats = 48 KB
#define B128_CHUNKS     (FLOATS_PER_TRAJ * 4 / 16)     // 3072
#define CHUNK_ITERS     (B128_CHUNKS / BLOCK)          // 12

__global__ __launch_bounds__(BLOCK)
void signature_kernel(const float* __restrict__ traj, float* __restrict__ out)
{
    __shared__ float s[FLOATS_PER_TRAJ];   // one full trajectory, 48 KB
    __shared__ float wred[8][BLOCK / 32];  // per-wave partials for 8 stats

    const int tid = threadIdx.x;

    // ---- Stage trajectory into LDS via CDNA5 async copies ----------------
    // flat-pointer low 32 bits are the
    // wave-relative LDS byte address (ISA 10.2 aperture mapping).
    const uint64_t gbase    = (uint64_t)(uintptr_t)(traj + (size_t)blockIdx.x * FLOATS_PER_TRAJ);
    const uint32_t lds_base = (uint32_t)(uintptr_t)(void*)s;

#pragma unroll
    for (int k = 0; k < CHUNK_ITERS; ++k) {
        uint32_t goff = (uint32_t)(k * BLOCK + tid) * 16u;   // coalesced B128 stream
        uint32_t loff = lds_base + goff;
        asm volatile("global_load_async_to_lds_b128 %0, %1, %2 th:TH_LOAD_NT"
                     :: "v"(loff), "v"(goff), "s"(gbase) : "memory");
    }
    asm volatile("s_wait_asynccnt 0" ::: "memory");
    __syncthreads();

    // ---- Per-thread partial sums over strided timesteps ------------------
    float sum_curv = 0.f, cntf = 0.f;
    float sum_dv = 0.f, sum_da = 0.f;
    float sum_sp = 0.f, sum_sp2 = 0.f;
    float sum_fo = 0.f, sum_fo2 = 0.f;

    for (int t = tid; t < TSTEPS; t += BLOCK) {
        const float* p = &s[t * NCH];
        float vx = p[2], vy = p[3];
        float ax = p[4], ay = p[5];

        float sp = sqrtf(vx * vx + vy * vy);
        sum_sp  += sp;  sum_sp2 += sp * sp;
        float fo = sqrtf(ax * ax + ay * ay);
        sum_fo  += fo;  sum_fo2 += fo * fo;

        if (t < TSTEPS - 1) {
            const float* q = p + NCH;
            sum_dv += fabsf(q[2] - vx) + fabsf(q[3] - vy);
            sum_da += fabsf(q[4] - ax) + fabsf(q[5] - ay);
        }
        if (t < TSTEPS - 2) {
            // v1 = delta[t+1], v2 = delta[t+2] (cumsum cancels in differences)
            float v1x = s[(t + 1) * NCH + 0], v1y = s[(t + 1) * NCH + 1];
            float v2x = s[(t + 2) * NCH + 0], v2y = s[(t + 2) * NCH + 1];
            float cross = v1x * v2y - v1y * v2x;
            float nrm = sqrtf(v1x * v1x + v1y * v1y) * sqrtf(v2x * v2x + v2y * v2y);
            if (nrm > 1e-6f) {
                sum_curv += fabsf(cross) / nrm;
                cntf     += 1.0f;
            }
        }
    }

    // ---- Wave32 shuffle reduction, then cross-wave via LDS ---------------
    float vals[8] = { sum_curv, cntf, sum_dv, sum_da, sum_sp, sum_sp2, sum_fo, sum_fo2 };
#pragma unroll
    for (int k = 0; k < 8; ++k) {
        float v = vals[k];
#pragma unroll
        for (int off = 16; off > 0; off >>= 1)
            v += __shfl_down(v, off, 32);   // wave32: 5 steps cover full wave
        vals[k] = v;
    }
    const int wave = tid >> 5;
    const int lane = tid & 31;
    if (lane == 0) {
#pragma unroll
        for (int k = 0; k < 8; ++k) wred[k][wave] = vals[k];
    }
    __syncthreads();

    // ---- Finalize (thread 0) ---------------------------------------------
    if (tid == 0) {
        float tot[8];
#pragma unroll
        for (int k = 0; k < 8; ++k) {
            float acc = 0.f;
#pragma unroll
            for (int w = 0; w < BLOCK / 32; ++w) acc += wred[k][w];
            tot[k] = acc;
        }

        const float inv_pairs = 1.0f / (2.0f * (TSTEPS - 1));  // mean over (T-1)x2
        const float inv_T     = 1.0f / (float)TSTEPS;

        float path_curv = (tot[1] > 0.f) ? (tot[0] / tot[1]) : 0.f;
        float vel_sm    = 1.0f / (1.0f + tot[2] * inv_pairs);
        float jerk      = tot[3] * inv_pairs;

        float msp  = tot[4] * inv_T;
        float vsp  = fmaxf(tot[5] * inv_T - msp * msp, 0.f);
        float rhythm = sqrtf(vsp) / (msp + 1e-6f);

        float mfo  = tot[6] * inv_T;
        float vfo  = fmaxf(tot[7] * inv_T - mfo * mfo, 0.f);
        float fmodu = sqrtf(vfo) / (mfo + 1e-6f);

        float* o = out + (size_t)blockIdx.x * 5;
        o[0] = path_curv;
        o[1] = vel_sm;
        o[2] = jerk;
        o[3] = rhythm;
        o[4] = fmodu;
    }
}

extern "C" void kernel_launch(void* const* d_in, const int* in_sizes, int n_in,
                              void* d_out, int out_size, void* d_ws, size_t ws_size,
                              hipStream_t stream)
{
    const float* traj = (const float*)d_in[0];
    float* out = (float*)d_out;
    const int ntraj = in_sizes[0] / FLOATS_PER_TRAJ;   // 4096
    signature_kernel<<<ntraj, BLOCK, 0, stream>>>(traj, out);
}